// GNNModel_23527830847722
// MI455X (gfx1250) — compile-verified
//
#include <hip/hip_runtime.h>
#include <hip/hip_bf16.h>
#include <stdint.h>

// Problem constants (fixed by the reference): x[T,N,D], edge_attr[T-1,E,D], edge_index[2,E]
#define T_DIM 64
#define N_DIM 10000
#define E_DIM 4096
#define D_DIM 128
#define EPB   256   // edges per block in the chain-eval kernel

// Native 16-byte vector (clang ext_vector) -> accepted by nontemporal builtins.
typedef float v4f __attribute__((ext_vector_type(4)));

// ---------------------------------------------------------------------------
// CDNA5 async global->LDS copy (gfx1250). Each active lane moves 16 bytes
// directly into LDS without touching VGPRs; tracked by ASYNCcnt.
// ---------------------------------------------------------------------------
__device__ __forceinline__ void async_load_b128_to_lds(uint32_t lds_byte_addr,
                                                       const void* gptr) {
  uint64_t ga = (uint64_t)(uintptr_t)gptr;
  asm volatile("global_load_async_to_lds_b128 %0, %1, off"
               :: "v"(lds_byte_addr), "v"(ga)
               : "memory");
}

__device__ __forceinline__ void wait_asynccnt0() {
  asm volatile("s_wait_asynccnt 0" ::: "memory");
}

// Flat pointer to LDS: low 32 bits of the flat address are the LDS byte offset
__device__ __forceinline__ uint32_t lds_addr_of(const void* p) {
  return (uint32_t)(uintptr_t)p;
}

// ---------------------------------------------------------------------------
// Kernel A: chain structure (shared by all timesteps).
//   pred[e]   = largest e' < e with dst[e'] == src[e], else -1
//   islast[e] = 1 iff no e' > e has dst[e'] == dst[e]
// dst[] is staged into LDS via async copies; scans run entirely out of LDS.
// ---------------------------------------------------------------------------
__global__ void gnn_preprocess(const int* __restrict__ edge_index,
                               int* __restrict__ pred,
                               int* __restrict__ islast) {
  __shared__ __align__(16) int dst_lds[E_DIM];
  const int* src = edge_index;          // edge_index[0, :]
  const int* dst = edge_index + E_DIM;  // edge_index[1, :]

  for (int i = threadIdx.x; i < E_DIM / 4; i += blockDim.x)
    async_load_b128_to_lds(lds_addr_of(&dst_lds[4 * i]), &dst[4 * i]);
  wait_asynccnt0();
  __syncthreads();

  const int e = blockIdx.x * blockDim.x + threadIdx.x;
  if (e < E_DIM) {
    const int s = src[e];
    int p = -1;
    for (int k = e - 1; k >= 0; --k) {
      if (dst_lds[k] == s) { p = k; break; }
    }
    pred[e] = p;

    const int de = dst_lds[e];
    int last = 1;
    for (int k = e + 1; k < E_DIM; ++k) {
      if (dst_lds[k] == de) { last = 0; break; }
    }
    islast[e] = last;
  }
}

// ---------------------------------------------------------------------------
// Kernel B: out[0] = x[0]; out[t] = x[t-1] for t >= 1 (rows untouched by edges
// keep the base value; written rows are overwritten by kernel C afterwards).
// Pure stream -> non-temporal on both sides so the 192 MB L2 is preserved for
// kernel C's reused x[t] root rows and index tables.
// ---------------------------------------------------------------------------
__global__ void gnn_shift_copy(const v4f* __restrict__ x4,
                               v4f* __restrict__ out4) {
  const int t  = blockIdx.y;
  const int st = (t == 0) ? 0 : (t - 1);
  const long nd4 = (long)N_DIM * D_DIM / 4;
  const long i = (long)blockIdx.x * blockDim.x + threadIdx.x;
  if (i < nd4) {
    const v4f v = __builtin_nontemporal_load(&x4[(long)st * nd4 + i]);
    __builtin_nontemporal_store(v, &out4[(long)t * nd4 + i]);
  }
}

// ---------------------------------------------------------------------------
// Kernel C: for every timestep t and every last-writer edge e, walk the pred
// chain (LDS pointer chase, wave-uniform), accumulate the edge_attr rows, and
// write out[t+1][dst[e]] = x[t][root] - sum(a).
// One wave per output row: 32 lanes x v4f = 128 contiguous floats.
// pred[], src[], islast[] staged into LDS via async global->LDS copies, so the
// dispatch loop + pointer chase touch no global memory at all.
// ---------------------------------------------------------------------------
__global__ void gnn_chain_eval(const float* __restrict__ x,
                               const float* __restrict__ ea,
                               const int* __restrict__ edge_index,
                               const int* __restrict__ pred,
                               const int* __restrict__ islast,
                               float* __restrict__ out) {
  __shared__ __align__(16) int pred_lds[E_DIM];
  __shared__ __align__(16) int src_lds[E_DIM];
  __shared__ __align__(16) int last_lds[E_DIM];
  const int* src = edge_index;
  const int* dst = edge_index + E_DIM;

  for (int i = threadIdx.x; i < E_DIM / 4; i += blockDim.x) {
    async_load_b128_to_lds(lds_addr_of(&pred_lds[4 * i]), &pred[4 * i]);
    async_load_b128_to_lds(lds_addr_of(&src_lds[4 * i]),  &src[4 * i]);
    async_load_b128_to_lds(lds_addr_of(&last_lds[4 * i]), &islast[4 * i]);
  }
  wait_asynccnt0();
  __syncthreads();

  const int t = blockIdx.y;  // 0 .. T-2
  const float* __restrict__ xt  = x  + (size_t)t * N_DIM * D_DIM;
  const float* __restrict__ eat = ea + (size_t)t * E_DIM * D_DIM;
  float* __restrict__ ot = out + (size_t)(t + 1) * (size_t)N_DIM * D_DIM;

  const int wave = threadIdx.x >> 5;   // 8 waves per block (wave32)
  const int lane = threadIdx.x & 31;   // lane covers columns [4*lane, 4*lane+4)
  const int e_begin = blockIdx.x * EPB;
  const int e_end   = e_begin + EPB;   // E_DIM divisible by EPB

  for (int e0 = e_begin + wave; e0 < e_end; e0 += 8) {
    if (!last_lds[e0]) continue;
    const int drow = dst[e0];          // output row; load early, broadcast

    v4f acc = {0.f, 0.f, 0.f, 0.f};
    int e = e0;
    int root;
    for (;;) {
      // edge_attr row is consumed exactly once along a chain -> non-temporal
      const v4f a = __builtin_nontemporal_load(
          (const v4f*)(eat + (size_t)e * D_DIM + 4 * lane));
      acc += a;
      const int p = pred_lds[e];       // wave-uniform LDS broadcast read
      if (p < 0) { root = src_lds[e]; break; }
      e = p;                           // pred(e) < e  -> guaranteed termination
    }
    // x[t] root rows may be shared by several chains -> keep regular-temporal
    const v4f xv = *(const v4f*)(xt + (size_t)root * D_DIM + 4 * lane);
    const v4f r = xv - acc;
    __builtin_nontemporal_store(
        r, (v4f*)(ot + (size_t)drow * D_DIM + 4 * lane));
  }
}

// ---------------------------------------------------------------------------
extern "C" void kernel_launch(void* const* d_in, const int* in_sizes, int n_in,
                              void* d_out, int out_size, void* d_ws, size_t ws_size,
                              hipStream_t stream) {
  const float* x  = (const float*)d_in[0];  // [T, N, D]
  const float* ea = (const float*)d_in[1];  // [T-1, E, D]
  const int*   ei = (const int*)d_in[2];    // [2, E]
  float* out = (float*)d_out;               // [T, N, D]

  int* pred   = (int*)d_ws;                 // E ints (16 KB)
  int* islast = pred + E_DIM;               // E ints (16 KB)

  // A: chain structure (shared across timesteps)
  gnn_preprocess<<<dim3(E_DIM / 256), dim3(256), 0, stream>>>(ei, pred, islast);

  // B: shifted base copy out[0]=x[0], out[t]=x[t-1]
  dim3 gB((unsigned)((N_DIM * D_DIM / 4 + 255) / 256), (unsigned)T_DIM);
  gnn_shift_copy<<<gB, dim3(256), 0, stream>>>((const v4f*)x, (v4f*)out);

  // C: evaluate all last-writer rows for all 63 timesteps in parallel
  dim3 gC((unsigned)(E_DIM / EPB), (unsigned)(T_DIM - 1));
  gnn_chain_eval<<<gC, dim3(256), 0, stream>>>(x, ea, ei, pred, islast, out);

  (void)in_sizes; (void)n_in; (void)out_size; (void)ws_size;
}